// GMAttendSoftmax2_28028956573866
// MI455X (gfx1250) — compile-verified
//
#include <hip/hip_runtime.h>
#include <hip/hip_bf16.h>
#include <math.h>

typedef __attribute__((ext_vector_type(2))) float v2f;
typedef __attribute__((ext_vector_type(4))) float v4f;
typedef __attribute__((ext_vector_type(8))) float v8f;

#define HIDDEN 512
#define KDIM   64
#define NB     8
#define NN     2048

// ---------------------------------------------------------------------------
// Kernel 1: projection.  out[row][c] = sum_h in[row][h]*W[c][h] + bias[c]
// One wave computes a 16-row x 64-col output tile (4 independent WMMA
// accumulator chains), so each x row is read exactly once.
// ---------------------------------------------------------------------------
__global__ __launch_bounds__(256) void proj_kernel(
    const float* __restrict__ x, const float* __restrict__ y,
    const float* __restrict__ Wq, const float* __restrict__ bq,
    const float* __restrict__ Wk, const float* __restrict__ bk,
    float* __restrict__ qout, float* __restrict__ kout)
{
    const int wave = threadIdx.x >> 5;
    const int lane = threadIdx.x & 31;
    const int col  = lane & 15;       // A-row / B-col index for this lane
    const int half = lane >> 4;       // K-pair selector
    const int rowbase = (blockIdx.x * 8 + wave) * 16;

    const float *in, *W, *bias; float* out;
    if (blockIdx.y == 0) { in = x; W = Wq; bias = bq; out = qout; }
    else                 { in = y; W = Wk; bias = bk; out = kout; }

    v8f c0 = {}, c1 = {}, c2 = {}, c3 = {};
    const float* arow = in + (size_t)(rowbase + col) * HIDDEN + 2 * half;
    const float* w0 = W + (size_t)(col +  0) * HIDDEN + 2 * half;
    const float* w1 = W + (size_t)(col + 16) * HIDDEN + 2 * half;
    const float* w2 = W + (size_t)(col + 32) * HIDDEN + 2 * half;
    const float* w3 = W + (size_t)(col + 48) * HIDDEN + 2 * half;

    for (int t = 0; t < HIDDEN / 4; ++t) {
        v2f a  = *(const v2f*)(arow + 4 * t);
        v2f b0 = *(const v2f*)(w0 + 4 * t);
        v2f b1 = *(const v2f*)(w1 + 4 * t);
        v2f b2 = *(const v2f*)(w2 + 4 * t);
        v2f b3 = *(const v2f*)(w3 + 4 * t);
        c0 = __builtin_amdgcn_wmma_f32_16x16x4_f32(false, a, false, b0, (short)0, c0, false, false);
        c1 = __builtin_amdgcn_wmma_f32_16x16x4_f32(false, a, false, b1, (short)0, c1, false, false);
        c2 = __builtin_amdgcn_wmma_f32_16x16x4_f32(false, a, false, b2, (short)0, c2, false, false);
        c3 = __builtin_amdgcn_wmma_f32_16x16x4_f32(false, a, false, b3, (short)0, c3, false, false);
    }

    // C/D layout: VGPR v -> row v (lanes 0-15) / row v+8 (lanes 16-31), col = lane%16
#pragma unroll
    for (int v = 0; v < 8; ++v) {
        const size_t row = (size_t)(rowbase + v + 8 * half);
        out[row * KDIM +  0 + col] = c0[v] + bias[ 0 + col];
        out[row * KDIM + 16 + col] = c1[v] + bias[16 + col];
        out[row * KDIM + 32 + col] = c2[v] + bias[32 + col];
        out[row * KDIM + 48 + col] = c3[v] + bias[48 + col];
    }
}

// ---------------------------------------------------------------------------
// Kernel 2: squared row norms of the q/k buffer (32768 rows x 64).
// One 32-lane wave per row: float2 load + xor-shuffle reduce.
// ---------------------------------------------------------------------------
__global__ __launch_bounds__(256) void norm_kernel(const float* __restrict__ qk,
                                                   float* __restrict__ sq)
{
    const int row  = blockIdx.x * 8 + (threadIdx.x >> 5);
    const int lane = threadIdx.x & 31;
    v2f d = *(const v2f*)(qk + (size_t)row * KDIM + lane * 2);
    float s = d.x * d.x + d.y * d.y;
#pragma unroll
    for (int m = 16; m >= 1; m >>= 1) s += __shfl_xor(s, m, 32);
    if (lane == 0) sq[row] = s;
}

// ---------------------------------------------------------------------------
// Kernel 3: fused dist-min + exp.  One wave owns 16 q rows of one batch,
// keeps the 16x64 A operand resident in 16 v2f registers, and streams all
// 2048 k rows in 32-wide tiles: TWO independent 16-step WMMA chains per
// iteration (c0 for cols mb..mb+15, c1 for mb+16..mb+31) so the XDL pipe
// always has an independent WMMA to issue.  Tiles are folded into running
// per-lane minima and never written to memory.
// w[b,n] = exp(-(min_m (|k_m|^2 - 2 q.k_m) + |q_n|^2)).
// ---------------------------------------------------------------------------
__global__ __launch_bounds__(256) void distmin_kernel(
    const float* __restrict__ q, const float* __restrict__ k,
    const float* __restrict__ sqq, const float* __restrict__ sqk,
    float* __restrict__ w)
{
    const int b    = blockIdx.y;
    const int wave = threadIdx.x >> 5;
    const int lane = threadIdx.x & 31;
    const int col  = lane & 15;
    const int half = lane >> 4;
    const int rowbase = (blockIdx.x * 8 + wave) * 16;

    const float* qb   = q   + (size_t)b * NN * KDIM;
    const float* kb   = k   + (size_t)b * NN * KDIM;
    const float* sqkb = sqk + (size_t)b * NN;

    v2f a[16];
#pragma unroll
    for (int t = 0; t < 16; ++t)
        a[t] = *(const v2f*)(qb + (size_t)(rowbase + col) * KDIM + 4 * t + 2 * half);

    v8f mn0, mn1;
#pragma unroll
    for (int v = 0; v < 8; ++v) { mn0[v] = 3.402823466e38f; mn1[v] = 3.402823466e38f; }

    for (int mb = 0; mb < NN; mb += 32) {
        v8f c0 = {}, c1 = {};
        const float* krow0 = kb + (size_t)(mb      + col) * KDIM + 2 * half;
        const float* krow1 = kb + (size_t)(mb + 16 + col) * KDIM + 2 * half;
#pragma unroll
        for (int t = 0; t < 16; ++t) {
            v2f bv0 = *(const v2f*)(krow0 + 4 * t);
            v2f bv1 = *(const v2f*)(krow1 + 4 * t);
            c0 = __builtin_amdgcn_wmma_f32_16x16x4_f32(false, a[t], false, bv0, (short)0, c0, false, false);
            c1 = __builtin_amdgcn_wmma_f32_16x16x4_f32(false, a[t], false, bv1, (short)0, c1, false, false);
        }
        const float sk0 = sqkb[mb + col];
        const float sk1 = sqkb[mb + 16 + col];
#pragma unroll
        for (int v = 0; v < 8; ++v) {
            mn0[v] = fminf(mn0[v], sk0 - 2.0f * c0[v]);
            mn1[v] = fminf(mn1[v], sk1 - 2.0f * c1[v]);
        }
    }

    // merge the two column streams, then min over 16 lanes (stays in-half)
    const float* sqqb = sqq + (size_t)b * NN;
    float*       wb   = w   + (size_t)b * NN;
#pragma unroll
    for (int v = 0; v < 8; ++v) {
        float m = fminf(mn0[v], mn1[v]);
#pragma unroll
        for (int s = 8; s >= 1; s >>= 1) m = fminf(m, __shfl_xor(m, s, 32));
        if (col == 0) {
            const int row = rowbase + v + 8 * half;
            wb[row] = __expf(-(m + sqqb[row]));
        }
    }
}

// ---------------------------------------------------------------------------
// Kernel 4: streaming outer product, out[b,n,m] = w[b,n]*w[b,m].
// 128 MiB of non-temporal 16B stores -> the bandwidth floor of the problem.
// Uses clang ext_vector (v4f) so __builtin_nontemporal_store accepts it.
// ---------------------------------------------------------------------------
__global__ __launch_bounds__(256) void outer_kernel(const float* __restrict__ w,
                                                    float* __restrict__ out)
{
    const int b = blockIdx.z;
    const int n = blockIdx.y;
    const int m4 = (blockIdx.x * blockDim.x + threadIdx.x) * 4;
    const float wn = w[(size_t)b * NN + n];
    const v4f v = *(const v4f*)(w + (size_t)b * NN + m4);
    v4f o = wn * v;
    __builtin_nontemporal_store(o, (v4f*)(out + ((size_t)b * NN + n) * NN + m4));
}

// ---------------------------------------------------------------------------
extern "C" void kernel_launch(void* const* d_in, const int* in_sizes, int n_in,
                              void* d_out, int out_size, void* d_ws, size_t ws_size,
                              hipStream_t stream)
{
    const float* x  = (const float*)d_in[0];
    const float* y  = (const float*)d_in[1];
    const float* Wq = (const float*)d_in[2];
    const float* bq = (const float*)d_in[3];
    const float* Wk = (const float*)d_in[4];
    const float* bk = (const float*)d_in[5];
    float* out = (float*)d_out;

    float* ws = (float*)d_ws;
    float* q  = ws;                         // 8*2048*64 = 1048576 floats
    float* kk = ws + 1048576;               // 1048576 floats (contiguous after q)
    float* sq = ws + 2097152;               // 32768 floats: q-row norms then k-row norms
    float* w  = ws + 2129920;               // 16384 floats

    // 1) projections: 16384 rows, 16 rows/wave, 8 waves/block -> 128 blocks; y: q vs k
    proj_kernel<<<dim3(128, 2), 256, 0, stream>>>(x, y, Wq, bq, Wk, bk, q, kk);
    // 2) row norms over q and k jointly (32768 rows, 8 rows/block)
    norm_kernel<<<4096, 256, 0, stream>>>(q, sq);
    // 3) fused min-dist + exp -> w  (128 row-groups/batch, 8 waves/block)
    distmin_kernel<<<dim3(16, NB), 256, 0, stream>>>(q, kk, sq, sq + 16384, w);
    // 4) outer product -> 128 MiB output
    outer_kernel<<<dim3(NN / (256 * 4), NN, NB), 256, 0, stream>>>(w, out);
}